// SoftPool3d_56977036149265
// MI455X (gfx1250) — compile-verified
//
#include <hip/hip_runtime.h>
#include <stdint.h>

// SoftPool3d fused single-pass kernel for MI455X (gfx1250).
// x: [4,64,16,112,112] f32  ->  out: [4,64,8,56,56] f32
// Memory-bound (231 MB min traffic @ 23.3 TB/s ~ 10us). One WG per
// (b, od, oh, w-half) tile; x staged to LDS once via CDNA5 async
// global->LDS b128 copies (inline asm, ASYNCcnt-tracked), then
// exp/channel-reduce + 2x2x2 windowed softmax-weighted average
// entirely out of LDS.

namespace {
constexpr int kB = 4, kC = 64, kD = 16, kH = 112, kW = 112;
constexpr int kOD = 8, kOH = 56, kOW = 56;
constexpr int kWHalf = 56;                 // input w span per block
constexpr int kOWh   = 28;                 // output w span per block
// input strides (floats)
constexpr int       kHS   = kW;            // 112
constexpr int       kDStr = kH * kW;       // 12544
constexpr int       kCS   = kD * kH * kW;  // 200704
constexpr long long kBS   = (long long)kC * kCS;
// output strides (floats)
constexpr int       kOHS = kOW;            // 56
constexpr int       kODS = kOH * kOW;      // 3136
constexpr int       kOCS = kOD * kODS;     // 25088
constexpr long long kOBS = (long long)kC * kOCS;

constexpr int kThreads      = 256;                       // 8 waves (wave32)
constexpr int kRows         = kC * 4;                    // 256 LDS rows (c,dz,hy)
constexpr int kChunksPerRow = kWHalf / 4;                // 14 x float4 per row
constexpr int kChunks       = kRows * kChunksPerRow;     // 3584
constexpr int kCopyIters    = kChunks / kThreads;        // 14
constexpr int kOutIters     = (kC * kOWh) / kThreads;    // 7
}  // namespace

typedef __attribute__((address_space(1))) const float GCF;
typedef __attribute__((address_space(3))) float SF;

// Per-lane async copy of 16 bytes global -> LDS (tracked by ASYNCcnt).
__device__ __forceinline__ void async_copy16(const float* gsrc, float* ldst) {
  SF* l3        = (SF*)ldst;                 // generic -> LDS addrspace
  GCF* g1       = (GCF*)gsrc;                // generic -> global addrspace
  unsigned loff = (unsigned)(uintptr_t)l3;   // 32-bit LDS byte offset
  asm volatile("global_load_async_to_lds_b128 %0, %1, off"
               :: "v"(loff), "v"(g1) : "memory");
}

__device__ __forceinline__ void wait_async_zero() {
  asm volatile("s_wait_asynccnt 0x0" ::: "memory");
}

__global__ __launch_bounds__(kThreads)
void softpool3d_fused(const float* __restrict__ x, float* __restrict__ out) {
  // 57344 + 896 + 112 bytes = 58352 B LDS per workgroup
  __shared__ __align__(16) float sx[kRows * kWHalf];  // [row = c*4 + dz*2 + hy][w]
  __shared__ float se[4 * kWHalf];                    // clamp(sum_c exp(x)) per (dz,hy,w)
  __shared__ float sden[kOWh];                        // 8-tap window sum of e per ow

  const int t  = threadIdx.x;
  unsigned gid = blockIdx.x;
  const int wb = gid & 1u;  gid >>= 1;
  const int oh = gid % kOH; gid /= kOH;
  const int od = gid % kOD; gid /= kOD;
  const int b  = (int)gid;

  // ---- Phase 1: async-stage 256 rows x 56 floats (56 KB) into LDS ----
  const float* gbase =
      x + (long long)b * kBS + (od * 2) * kDStr + (oh * 2) * kHS + wb * kWHalf;
#pragma unroll
  for (int it = 0; it < kCopyIters; ++it) {
    const int q  = t + it * kThreads;        // 16B chunk id
    const int r  = q / kChunksPerRow;        // row 0..255
    const int k  = q - r * kChunksPerRow;    // float4 within row 0..13
    const int c  = r >> 2;
    const int dz = (r >> 1) & 1;
    const int hy = r & 1;
    const float* gsrc =
        gbase + (long long)c * kCS + dz * kDStr + hy * kHS + k * 4;
    async_copy16(gsrc, &sx[r * kWHalf + k * 4]);
  }
  wait_async_zero();
  __syncthreads();

  // ---- Phase 2: e[dz,hy,w] = clamp(sum_c exp(x), 0) ----
  if (t < 4 * kWHalf) {                      // 224 active threads
    const int r4 = t / kWHalf;               // dz*2 + hy
    const int w  = t - r4 * kWHalf;
    float s = 0.f;
#pragma unroll 4
    for (int c = 0; c < kC; ++c)
      s += __expf(sx[(c * 4 + r4) * kWHalf + w]);
    se[r4 * kWHalf + w] = fmaxf(s, 0.f);
  }
  __syncthreads();

  // ---- Phase 2.5: den[ow] = sum of e over the 2x2x2 window ----
  if (t < kOWh) {
    const int w2 = t * 2;
    float d = 0.f;
#pragma unroll
    for (int r4 = 0; r4 < 4; ++r4)
      d += se[r4 * kWHalf + w2] + se[r4 * kWHalf + w2 + 1];
    sden[t] = d;
  }
  __syncthreads();

  // ---- Phase 3: out[c,ow] = relu( (sum_window x*e) / den ) ----
  float* obase =
      out + (long long)b * kOBS + od * kODS + oh * kOHS + wb * kOWh;
#pragma unroll
  for (int it = 0; it < kOutIters; ++it) {
    const int o  = t + it * kThreads;        // 0..1791
    const int c  = o / kOWh;
    const int ow = o - c * kOWh;
    const int w2 = ow * 2;
    float num = 0.f;
#pragma unroll
    for (int r4 = 0; r4 < 4; ++r4) {
      const float* row = &sx[(c * 4 + r4) * kWHalf + w2];
      num += row[0] * se[r4 * kWHalf + w2] +
             row[1] * se[r4 * kWHalf + w2 + 1];
    }
    const float v = num / sden[ow];
    obase[(long long)c * kOCS + ow] = fmaxf(v, 0.f);
  }
}

extern "C" void kernel_launch(void* const* d_in, const int* in_sizes, int n_in,
                              void* d_out, int out_size, void* d_ws, size_t ws_size,
                              hipStream_t stream) {
  (void)in_sizes; (void)n_in; (void)out_size; (void)d_ws; (void)ws_size;
  const float* x = (const float*)d_in[0];
  float* out     = (float*)d_out;
  const int grid = kB * kOD * kOH * 2;  // 3584 workgroups
  softpool3d_fused<<<grid, kThreads, 0, stream>>>(x, out);
}